// BasicBlock_52372831207986
// MI455X (gfx1250) — compile-verified
//
#include <hip/hip_runtime.h>

typedef __attribute__((ext_vector_type(2))) float v2f;
typedef __attribute__((ext_vector_type(8))) float v8f;

#define C_CH   128
#define HW     3136      // 56*56
#define KTOT   1152      // 9*128, K ordered as k = r*128 + c  (r = dy*3+dx)
#define M_TILE 64
#define K_CHNK 32
#define NCHUNK (KTOT / K_CHNK)   // 36; 128%32==0 -> each chunk has a single r
#define LDA    34        // A row stride (floats): even (8B v2f), banks spread
#define LDB    36        // B row stride (floats): 16B-aligned rows, banks spread
#define LDC    68        // C-exchange row stride: 16B-aligned, banks spread
#define A_SZ   (M_TILE * LDA)    // 2176
#define B_SZ   (C_CH * LDB)      // 4608
#define SMEM_F (2 * A_SZ + 2 * B_SZ)   // 13568 floats = 54.3 KB (lC overlays)

// ---------------------------------------------------------------------------
// Fold BN into conv weights, transposing K to kernel-position-major:
//   wt[o*1152 + r*128 + c] = w[o*1152 + c*9 + r] * gamma*rsqrt(var+eps)
//   bias[o]                = beta - mean*scale
// ---------------------------------------------------------------------------
__global__ void fold_bn_kernel(const float* __restrict__ w1, const float* __restrict__ w2,
                               const float* __restrict__ g1, const float* __restrict__ b1,
                               const float* __restrict__ m1, const float* __restrict__ v1,
                               const float* __restrict__ g2, const float* __restrict__ b2,
                               const float* __restrict__ m2, const float* __restrict__ v2,
                               float* __restrict__ wt1, float* __restrict__ wt2,
                               float* __restrict__ bias1, float* __restrict__ bias2) {
    int idx = blockIdx.x * blockDim.x + threadIdx.x;
    if (idx < C_CH * KTOT) {
        int o   = idx / KTOT;
        int rem = idx - o * KTOT;
        int r   = rem >> 7;          // 0..8
        int c   = rem & 127;
        int src = o * KTOT + c * 9 + r;
        float s1 = g1[o] * rsqrtf(v1[o] + 1e-5f);
        float s2 = g2[o] * rsqrtf(v2[o] + 1e-5f);
        wt1[idx] = w1[src] * s1;
        wt2[idx] = w2[src] * s2;
    }
    if (idx < C_CH) {
        float s1 = g1[idx] * rsqrtf(v1[idx] + 1e-5f);
        float s2 = g2[idx] * rsqrtf(v2[idx] + 1e-5f);
        bias1[idx] = b1[idx] - m1[idx] * s1;
        bias2[idx] = b2[idx] - m2[idx] * s2;
    }
}

// ---------------------------------------------------------------------------
// Implicit-GEMM 3x3 conv (pad 1, stride 1) + bias + [residual] + ReLU.
// Block: 256 threads = 8 wave32, computes 64(M) x 128(N) output tile.
// Wave w: M sub-tile (w&3)*16, N sub-tile (w>>2)*64 -> 4 v8f accumulators.
// Math: v_wmma_f32_16x16x4_f32; K double-buffered through LDS, 32 per chunk.
// Epilogue: LDS transpose to [o][m] -> fully coalesced b128 stores.
// ---------------------------------------------------------------------------
template <bool FUSE_RES>
__global__ __launch_bounds__(256, 2)
void conv3x3_bn_wmma(const float* __restrict__ in,    // [64,128,56,56]
                     const float* __restrict__ wt,    // folded, [o][r*128+c]
                     const float* __restrict__ bias,  // [128]
                     const float* __restrict__ res,   // residual (or null)
                     float* __restrict__ out) {       // [64,128,56,56]
    __shared__ float smem[SMEM_F];
    float* lA0 = smem;
    float* lA1 = smem + A_SZ;
    float* lB0 = smem + 2 * A_SZ;
    float* lB1 = smem + 2 * A_SZ + B_SZ;
    float* lC  = smem;                       // overlays staging after main loop

    const int tid    = threadIdx.x;
    const int m_base = blockIdx.x * M_TILE;

    const int lane  = tid & 31;
    const int wid   = tid >> 5;
    const int m_off = (wid & 3) << 4;    // 0,16,32,48
    const int n_off = (wid >> 2) << 6;   // 0,64
    const int lm    = lane & 15;
    const int hi    = lane >> 4;         // lane half (selects K 0/1 vs 2/3)

    // --- A-tile loader: thread -> (m_local fixed, kk = a_k0 + 4i, i=0..7) ---
    const int a_ml  = tid & 63;
    const int a_k0  = tid >> 6;          // 0..3
    const int am    = m_base + a_ml;
    const int an    = am / HW;           // image index (uniform: 3136%64==0)
    const int arem  = am - an * HW;      // h*56+w
    const int ah    = arem / 56;
    const int aw    = arem - ah * 56;
    const float* in_n  = in + an * (C_CH * HW);
    const int    a_dst = a_ml * LDA + a_k0;

    // --- B-tile loader: thread -> o = tid>>1, 16 contiguous k ---
    const int b_o   = tid >> 1;
    const int b_kk  = (tid & 1) << 4;
    const float* wt_p  = wt + b_o * KTOT + b_kk;
    const int    b_dst = b_o * LDB + b_kk;

    // --- per-wave fragment row bases ---
    const int a_row = (m_off + lm) * LDA;
    const int b_row = (n_off + lm) * LDB;

    v8f acc[4];
#pragma unroll
    for (int t = 0; t < 4; ++t) acc[t] = (v8f){};

    // Stage one 32-wide K chunk into LDS buffer `buf`.
    // k = r*128 + c: the whole chunk shares one (dy,dx) kernel position.
    auto stage = [&](int k0, int buf) {
        const int r  = k0 >> 7;          // 0..8
        const int c0 = (k0 & 127) + a_k0;
        const int dy = r / 3;
        const int dx = r - dy * 3;
        const int hh = ah + dy - 1;
        const int ww = aw + dx - 1;
        const bool ok = (((unsigned)hh < 56u) & ((unsigned)ww < 56u));
        float* dst = (buf ? lA1 : lA0) + a_dst;
        if (ok) {
            const float* src = in_n + c0 * HW + hh * 56 + ww;
#pragma unroll
            for (int i = 0; i < 8; ++i) dst[i * 4] = src[i * 4 * HW];
        } else {
#pragma unroll
            for (int i = 0; i < 8; ++i) dst[i * 4] = 0.0f;
        }
        const float* p = wt_p + k0;
        float* db = (buf ? lB1 : lB0) + b_dst;
#pragma unroll
        for (int j = 0; j < 16; ++j) db[j] = p[j];
        if (k0 + K_CHNK < KTOT)
            __builtin_prefetch(p + K_CHNK, 0, 3);  // global_prefetch_b8, near
    };

    stage(0, 0);
    __syncthreads();

    for (int i = 0; i < NCHUNK; ++i) {
        const int buf = i & 1;
        if (i + 1 < NCHUNK) stage((i + 1) * K_CHNK, buf ^ 1);

        const float* A = buf ? lA1 : lA0;
        const float* B = buf ? lB1 : lB0;
#pragma unroll
        for (int ks = 0; ks < 8; ++ks) {
            const int ka = (ks << 2) + (hi << 1);           // even -> 8B aligned
            v2f a  = *(const v2f*)&A[a_row + ka];
            v2f b0 = *(const v2f*)&B[b_row + ka];
            v2f b1 = *(const v2f*)&B[b_row + 16 * LDB + ka];
            v2f b2 = *(const v2f*)&B[b_row + 32 * LDB + ka];
            v2f b3 = *(const v2f*)&B[b_row + 48 * LDB + ka];
            acc[0] = __builtin_amdgcn_wmma_f32_16x16x4_f32(false, a, false, b0, (short)0, acc[0], false, false);
            acc[1] = __builtin_amdgcn_wmma_f32_16x16x4_f32(false, a, false, b1, (short)0, acc[1], false, false);
            acc[2] = __builtin_amdgcn_wmma_f32_16x16x4_f32(false, a, false, b2, (short)0, acc[2], false, false);
            acc[3] = __builtin_amdgcn_wmma_f32_16x16x4_f32(false, a, false, b3, (short)0, acc[3], false, false);
        }
        __syncthreads();
    }

    // ---- epilogue stage 1: dump accumulators to LDS as [o][m] ----
    // C/D layout: VGPR v, lanes 0-15 -> (M=v, N=lane); lanes 16-31 -> (M=v+8).
    const int c_m = m_off + (hi << 3);
#pragma unroll
    for (int t = 0; t < 4; ++t) {
        float* row = &lC[(n_off + (t << 4) + lm) * LDC + c_m];
#pragma unroll
        for (int v = 0; v < 8; ++v) row[v] = acc[t][v];
    }
    __syncthreads();

    // ---- epilogue stage 2: coalesced bias + (residual) + ReLU stores ----
    // Blocks never straddle images, so each (block, o) run is 64 contiguous
    // floats in NCHW memory. Thread t: channel t>>1, half (t&1) of the 64.
    {
        const int o     = tid >> 1;
        const int ms    = (tid & 1) << 5;
        const int n_img = m_base / HW;
        const int rem0  = m_base - n_img * HW;
        const float bv  = bias[o];
        const float* srcc = &lC[o * LDC + ms];
        const int    gidx = (n_img * C_CH + o) * HW + rem0 + ms;
        float*       dst  = out + gidx;
        const float* rp   = FUSE_RES ? (res + gidx) : nullptr;
#pragma unroll
        for (int j = 0; j < 32; ++j) {
            float val = srcc[j] + bv;
            if (FUSE_RES) val += rp[j];
            dst[j] = fmaxf(val, 0.0f);
        }
    }
}

// ---------------------------------------------------------------------------
extern "C" void kernel_launch(void* const* d_in, const int* in_sizes, int n_in,
                              void* d_out, int out_size, void* d_ws, size_t ws_size,
                              hipStream_t stream) {
    (void)in_sizes; (void)n_in; (void)out_size; (void)ws_size;
    const float* x  = (const float*)d_in[0];
    const float* w1 = (const float*)d_in[1];
    const float* w2 = (const float*)d_in[2];
    const float* g1 = (const float*)d_in[3];
    const float* b1 = (const float*)d_in[4];
    const float* m1 = (const float*)d_in[5];
    const float* v1 = (const float*)d_in[6];
    const float* g2 = (const float*)d_in[7];
    const float* b2 = (const float*)d_in[8];
    const float* m2 = (const float*)d_in[9];
    const float* v2 = (const float*)d_in[10];

    float* ws    = (float*)d_ws;
    float* wt1   = ws;                          // 147456
    float* wt2   = wt1 + C_CH * KTOT;           // 147456
    float* bias1 = wt2 + C_CH * KTOT;           // 128
    float* bias2 = bias1 + C_CH;                // 128
    float* y1    = bias2 + C_CH;                // 64*128*3136 intermediate

    fold_bn_kernel<<<(C_CH * KTOT + 255) / 256, 256, 0, stream>>>(
        w1, w2, g1, b1, m1, v1, g2, b2, m2, v2, wt1, wt2, bias1, bias2);

    const int nblocks = (64 * HW) / M_TILE;     // 3136
    conv3x3_bn_wmma<false><<<nblocks, 256, 0, stream>>>(x,  wt1, bias1, nullptr, y1);
    conv3x3_bn_wmma<true ><<<nblocks, 256, 0, stream>>>(y1, wt2, bias2, x, (float*)d_out);
}